// ERecomposition_824633721644
// MI455X (gfx1250) — compile-verified
//
#include <hip/hip_runtime.h>

typedef __attribute__((ext_vector_type(2))) float v2f;
typedef __attribute__((ext_vector_type(8))) float v8f;

#define N_ATOMS   64
#define N_PAIRS   2016          // 64*63/2
#define NTHREADS  320           // 10 wave32 waves: one per lower-tri 16x16 tile

// One block per batch element.
// Gram matrix G = C*C^T (C: 64x3) via V_WMMA_F32_16X16X4_F32 tiles (K padded to 4).
// d2(i,j) = n2[i] + n2[j] - 2*G[i][j];  rr = rsqrt(d2)
// out[b]  = (sum rr * fe[p]) * rsqrt(sum rr^2)
__global__ __launch_bounds__(NTHREADS) void erecomp_kernel(
    const float* __restrict__ coords,    // [B, 64, 3]
    const float* __restrict__ decompFE,  // [B, 2016]
    float* __restrict__ out)             // [B]
{
    // rows: 0=x, 1=y, 2=z, 3=zeros (K=3 pad source) -> operand select is pure address math
    __shared__ __align__(16) float sC[4][N_ATOMS];
    __shared__ __align__(16) float sN2[N_ATOMS];
    __shared__ __align__(16) float sFE[N_PAIRS];
    __shared__ float sRed1[10], sRed2[10];

    const int b    = blockIdx.x;
    const int tid  = threadIdx.x;
    const int lane = tid & 31;
    const int wave = tid >> 5;          // 0..9

    // ---- stage coords (SoA) + per-atom squared norms into LDS ----
    const float* cb = coords + (size_t)b * (N_ATOMS * 3);
    if (tid < N_ATOMS) {
        float x = cb[tid * 3 + 0];       // 12 contiguous bytes/thread
        float y = cb[tid * 3 + 1];
        float z = cb[tid * 3 + 2];
        sC[0][tid] = x; sC[1][tid] = y; sC[2][tid] = z; sC[3][tid] = 0.0f;
        sN2[tid] = x * x + y * y + z * z;
    }
    // ---- stream this batch's decompFE row into LDS: 504 float4 = b128 coalesced ----
    {
        const float4* fb4  = (const float4*)(decompFE + (size_t)b * N_PAIRS);
        float4*       sFE4 = (float4*)sFE;
        #pragma unroll 2
        for (int idx = tid; idx < N_PAIRS / 4; idx += NTHREADS) sFE4[idx] = fb4[idx];
    }
    __syncthreads();

    // ---- tile assignment: wave w -> (ti, tj) with ti >= tj, row-major lower tri ----
    const int ti = (wave >= 6) ? 3 : (wave >= 3) ? 2 : (wave >= 1) ? 1 : 0;
    const int tj = wave - (ti * (ti + 1)) / 2;

    const int laneM = lane & 15;
    const int hi    = lane >> 4;        // 0: K=0/1 half, 1: K=2/3 half
    const int h2    = hi * 2;           // row pair base in sC: {x,y} or {z,0}
    const int rowAtom = ti * 16 + laneM;
    const int colAtom = tj * 16 + laneM;

    // A (16x4 f32): VGPR0 = {K0 | K2}, VGPR1 = {K1 | K3}; K3 = 0 (from sC row 3)
    v2f a, bm;
    a.x  = sC[h2    ][rowAtom];
    a.y  = sC[h2 + 1][rowAtom];
    // B (4x16 f32), mirrored layout over columns
    bm.x = sC[h2    ][colAtom];
    bm.y = sC[h2 + 1][colAtom];

    v8f acc = {0.f, 0.f, 0.f, 0.f, 0.f, 0.f, 0.f, 0.f};
    // emits v_wmma_f32_16x16x4_f32 (EXEC all-ones at this point)
    acc = __builtin_amdgcn_wmma_f32_16x16x4_f32(
        /*neg_a=*/false, a, /*neg_b=*/false, bm,
        /*c_mod=*/(short)0, acc, /*reuse_a=*/false, /*reuse_b=*/false);

    // ---- branch-free post-processing of the 16x16 f32 C/D layout ----
    // VGPR v holds M=v (lanes 0-15) and M=v+8 (lanes 16-31); N = laneM.
    const int   j   = tj * 16 + laneM;          // global column
    const float n2j = sN2[j];
    const int   i0  = ti * 16 + hi * 8;         // first global row of this lane's 8 elems

    // 8 consecutive sN2 values via two aligned ds_load_b128
    const float4* n2v = (const float4*)&sN2[i0];
    const float4  n2a = n2v[0];
    const float4  n2b = n2v[1];
    const float   n2i[8] = { n2a.x, n2a.y, n2a.z, n2a.w, n2b.x, n2b.y, n2b.z, n2b.w };

    float s1 = 0.0f, s2 = 0.0f;
    int   tri = (i0 * (i0 - 1)) >> 1;           // triangular base tri(i0), one mul total
#pragma unroll
    for (int v = 0; v < 8; ++v) {
        const int i  = i0 + v;
        float d2 = n2i[v] + n2j - 2.0f * acc[v];
        float rq = __builtin_amdgcn_rsqf(d2);       // bare v_rsq_f32, unconditional
        float rr = (i > j) ? rq : 0.0f;             // single v_cndmask
        int   p  = tri + j;                         // packed tril pair index
        p = (p < N_PAIRS - 1) ? p : (N_PAIRS - 1);  // v_min; invalid lanes read in-bounds
        s1 += rr * rr;                              // sum 1/r^2 (0 when invalid)
        s2 += rr * sFE[p];                          // rr==0 kills invalid term
        tri += i;                                   // tri(i+1) = tri(i) + i
    }

    // ---- wave32 shuffle reduction, then cross-wave via LDS ----
#pragma unroll
    for (int off = 16; off > 0; off >>= 1) {
        s1 += __shfl_down(s1, off, 32);
        s2 += __shfl_down(s2, off, 32);
    }
    if (lane == 0) { sRed1[wave] = s1; sRed2[wave] = s2; }
    __syncthreads();

    if (tid == 0) {
        float t1 = 0.0f, t2 = 0.0f;
#pragma unroll
        for (int w = 0; w < 10; ++w) { t1 += sRed1[w]; t2 += sRed2[w]; }
        out[b] = t2 * __builtin_amdgcn_rsqf(t1);   // S2 / sqrt(S1)
    }
}

extern "C" void kernel_launch(void* const* d_in, const int* in_sizes, int n_in,
                              void* d_out, int out_size, void* d_ws, size_t ws_size,
                              hipStream_t stream) {
    const float* coords   = (const float*)d_in[0];   // [B,64,3] f32
    const float* decompFE = (const float*)d_in[1];   // [B,2016] f32
    float* out = (float*)d_out;                      // [B,1] f32
    const int B = in_sizes[0] / (N_ATOMS * 3);       // 8192
    (void)n_in; (void)out_size; (void)d_ws; (void)ws_size;
    erecomp_kernel<<<B, NTHREADS, 0, stream>>>(coords, decompFE, out);
}